// gpn_layer_46076409152049
// MI455X (gfx1250) — compile-verified
//
#include <hip/hip_runtime.h>
#include <hip/hip_bf16.h>
#include <math.h>

typedef __attribute__((ext_vector_type(2))) float v2f;
typedef __attribute__((ext_vector_type(8))) float v8f;

#define B_   32
#define M_   64
#define N_   36
#define L_   1024
#define HID_ 512
#define NMAX_ 128
#define G_   (B_ * M_)      // 2048
#define RO_  (2 * L_)       // 2048 read_out width

// ---------------------------------------------------------------------------
// Kernel A: per-g fused  gather(att) -> pool(36x36) @ att(36x1024) -> row max
// and mean over N=36 -> read_out[g][0:1024]=max, [1024:2048]=mean.
// 1 block per g, 1024 threads = 32 waves; wave w owns columns [w*32, w*32+32).
// WMMA: f32 16x16x4, M padded 36->48 (3 tiles), K=36=9*4 exact, N: 2 tiles.
// ---------------------------------------------------------------------------
__global__ __launch_bounds__(1024) void gpn_pool_gemm_reduce(
    const int* __restrict__ obj_ind,     // (32,64,36)
    const float* __restrict__ pool_mtx,  // (64,32,36,36)  (M,b,N,N)
    const float* __restrict__ att,       // (32,128,1024)
    float* __restrict__ read_out)        // (2048,2048)
{
    __shared__ float poolLds[48][36];    // zero-padded rows 36..47
    __shared__ int   rowbase[36];        // element offsets into att

    const int g   = blockIdx.x;
    const int bi  = g / M_;
    const int mi  = g % M_;
    const int tid = threadIdx.x;

    for (int i = tid; i < 48 * 36; i += 1024) {
        int r = i / 36, c = i % 36;
        float v = 0.0f;
        if (r < 36) v = pool_mtx[(((size_t)mi * B_ + bi) * 36 + r) * 36 + c];
        poolLds[r][c] = v;
    }
    if (tid < 36) {
        int o = obj_ind[((size_t)bi * M_ + mi) * 36 + tid];
        rowbase[tid] = (bi * NMAX_ + o) * L_;
    }
    __syncthreads();

    const int wave = tid >> 5;
    const int lane = tid & 31;
    const int ln16 = lane & 15;
    const int hi   = lane >> 4;
    const int c0   = wave * 32;

    v8f acc[3][2];
    for (int i = 0; i < 3; ++i)
        for (int j = 0; j < 2; ++j)
            acc[i][j] = (v8f)0.0f;

    for (int k = 0; k < 9; ++k) {
        const int mbase = k * 4 + hi * 2;
        v2f a0, a1, a2, b0, b1;
        a0.x = poolLds[ 0 + ln16][mbase];  a0.y = poolLds[ 0 + ln16][mbase + 1];
        a1.x = poolLds[16 + ln16][mbase];  a1.y = poolLds[16 + ln16][mbase + 1];
        a2.x = poolLds[32 + ln16][mbase];  a2.y = poolLds[32 + ln16][mbase + 1];
        const float* r0 = att + (size_t)rowbase[mbase];
        const float* r1 = att + (size_t)rowbase[mbase + 1];
        b0.x = r0[c0 + ln16];       b0.y = r1[c0 + ln16];
        b1.x = r0[c0 + 16 + ln16];  b1.y = r1[c0 + 16 + ln16];

        acc[0][0] = __builtin_amdgcn_wmma_f32_16x16x4_f32(false, a0, false, b0, (short)0, acc[0][0], false, false);
        acc[0][1] = __builtin_amdgcn_wmma_f32_16x16x4_f32(false, a0, false, b1, (short)0, acc[0][1], false, false);
        acc[1][0] = __builtin_amdgcn_wmma_f32_16x16x4_f32(false, a1, false, b0, (short)0, acc[1][0], false, false);
        acc[1][1] = __builtin_amdgcn_wmma_f32_16x16x4_f32(false, a1, false, b1, (short)0, acc[1][1], false, false);
        acc[2][0] = __builtin_amdgcn_wmma_f32_16x16x4_f32(false, a2, false, b0, (short)0, acc[2][0], false, false);
        acc[2][1] = __builtin_amdgcn_wmma_f32_16x16x4_f32(false, a2, false, b1, (short)0, acc[2][1], false, false);
    }

    // Row reduction over n=0..35.  C layout: vgpr r = rows {tile*16 + hi*8 + r},
    // column = tile_col_base + ln16.  Tile2 valid rows: 32..35 (vgpr 0..3, hi==0).
    for (int nt = 0; nt < 2; ++nt) {
        float mx = -INFINITY, sm = 0.0f;
        for (int t = 0; t < 2; ++t)
            for (int r = 0; r < 8; ++r) {
                float v = acc[t][nt][r];
                mx = fmaxf(mx, v);
                sm += v;
            }
        for (int r = 0; r < 4; ++r) {
            float v = acc[2][nt][r];
            if (hi == 0) mx = fmaxf(mx, v);   // rows 32..35 only
            sm += v;                          // padded rows are exactly 0
        }
        float mx2 = __shfl_xor(mx, 16, 32);
        float sm2 = __shfl_xor(sm, 16, 32);
        mx = fmaxf(mx, mx2);
        sm += sm2;
        if (hi == 0) {
            int col = c0 + nt * 16 + ln16;
            read_out[(size_t)g * RO_ + col]      = mx;
            read_out[(size_t)g * RO_ + L_ + col] = sm * (1.0f / 36.0f);
        }
    }
}

// ---------------------------------------------------------------------------
// Kernel B1: h = relu(read_out @ W1 + b1)   (2048x2048)@(2048x512)
// Block = 256 thr (8 waves as 2x4). Block tile 32 rows x 128 cols.
// Grid: x = row tile (64), y = col tile (4).
// ---------------------------------------------------------------------------
__global__ __launch_bounds__(256) void gpn_hidden_gemm(
    const float* __restrict__ read_out,  // (2048,2048)
    const float* __restrict__ W1,        // (2048,512)
    const float* __restrict__ b1,        // (512)
    float* __restrict__ h)               // (2048,512)
{
    const int tid  = threadIdx.x;
    const int wave = tid >> 5;
    const int lane = tid & 31;
    const int ln16 = lane & 15;
    const int hi   = lane >> 4;
    const int wm   = wave >> 2;          // 0..1
    const int wn   = wave & 3;           // 0..3
    const int r0   = blockIdx.x * 32 + wm * 16;
    const int c0   = blockIdx.y * 128 + wn * 32;

    v8f acc[2];
    acc[0] = (v8f)0.0f;
    acc[1] = (v8f)0.0f;

    for (int kk = 0; kk < RO_; kk += 4) {
        const int mbase = kk + hi * 2;
        v2f a, b0, b1v;
        const float* ar = read_out + (size_t)(r0 + ln16) * RO_;
        a.x  = ar[mbase];
        a.y  = ar[mbase + 1];
        b0.x  = W1[(size_t)mbase * HID_ + c0 + ln16];
        b0.y  = W1[(size_t)(mbase + 1) * HID_ + c0 + ln16];
        b1v.x = W1[(size_t)mbase * HID_ + c0 + 16 + ln16];
        b1v.y = W1[(size_t)(mbase + 1) * HID_ + c0 + 16 + ln16];
        acc[0] = __builtin_amdgcn_wmma_f32_16x16x4_f32(false, a, false, b0,  (short)0, acc[0], false, false);
        acc[1] = __builtin_amdgcn_wmma_f32_16x16x4_f32(false, a, false, b1v, (short)0, acc[1], false, false);
    }

    for (int nt = 0; nt < 2; ++nt) {
        int col = c0 + nt * 16 + ln16;
        float bias = b1[col];
        for (int r = 0; r < 8; ++r) {
            int row = r0 + hi * 8 + r;
            float v = acc[nt][r] + bias;
            h[(size_t)row * HID_ + col] = fmaxf(v, 0.0f);
        }
    }
}

// ---------------------------------------------------------------------------
// Kernel B2: score[g] = sigmoid(h[g,:] . W2 + b2).  One wave per row.
// ---------------------------------------------------------------------------
__global__ __launch_bounds__(256) void gpn_score(
    const float* __restrict__ h, const float* __restrict__ W2,
    const float* __restrict__ b2, float* __restrict__ score)
{
    const int wave = threadIdx.x >> 5;
    const int lane = threadIdx.x & 31;
    const int row  = blockIdx.x * 8 + wave;
    const float* hr = h + (size_t)row * HID_;
    float p = 0.0f;
    for (int t = 0; t < 16; ++t) {
        int j = lane * 16 + t;
        p += hr[j] * W2[j];
    }
    for (int off = 16; off >= 1; off >>= 1)
        p += __shfl_xor(p, off, 32);
    if (lane == 0)
        score[row] = 1.0f / (1.0f + expf(-(p + b2[0])));
}

// ---------------------------------------------------------------------------
// Kernel B3: per-batch argmax over M=64 scores (first-max tie-break), write
// sub_max_ind (as float), att_masks_out, and stash mi in ws.
// One wave (32 threads) per batch.
// ---------------------------------------------------------------------------
__global__ __launch_bounds__(32) void gpn_argmax(
    const float* __restrict__ score,      // (2048,)
    const float* __restrict__ att_masks,  // (32,64,36)
    float* __restrict__ out_submax,       // (32,)
    float* __restrict__ out_masks,        // (32,36)
    int* __restrict__ ws_mi)              // (32,)
{
    const int bi   = blockIdx.x;
    const int lane = threadIdx.x;
    float v = score[bi * M_ + lane];
    int   i = lane;
    float v1 = score[bi * M_ + 32 + lane];
    if (v1 > v) { v = v1; i = 32 + lane; }
    for (int off = 16; off >= 1; off >>= 1) {
        float ov = __shfl_xor(v, off, 32);
        int   oi = __shfl_xor(i, off, 32);
        if (ov > v || (ov == v && oi < i)) { v = ov; i = oi; }
    }
    if (lane == 0) {
        ws_mi[bi]      = i;
        out_submax[bi] = (float)i;
    }
    for (int j = lane; j < N_; j += 32)
        out_masks[bi * N_ + j] = att_masks[((size_t)bi * M_ + i) * N_ + j];
}

// ---------------------------------------------------------------------------
// Kernel C1: h2 = sub_read_out @ Wp1 + bp1   (32 x 512), K = 2048
// ---------------------------------------------------------------------------
__global__ __launch_bounds__(256) void gpn_fc1(
    const float* __restrict__ read_out, const int* __restrict__ ws_mi,
    const float* __restrict__ Wp1, const float* __restrict__ bp1,
    float* __restrict__ h2)
{
    const int idx = blockIdx.x * 256 + threadIdx.x;   // 0..16383
    const int bi  = idx >> 9;
    const int j   = idx & 511;
    const float* rr = read_out + (size_t)(bi * M_ + ws_mi[bi]) * RO_;
    float s = bp1[j];
    for (int k = 0; k < RO_; ++k)
        s += rr[k] * Wp1[(size_t)k * HID_ + j];
    h2[idx] = s;
}

// ---------------------------------------------------------------------------
// Kernel C2: fc_feats_out = h2 @ Wp2 + bp2   (32 x 2048), K = 512
// ---------------------------------------------------------------------------
__global__ __launch_bounds__(256) void gpn_fc2(
    const float* __restrict__ h2, const float* __restrict__ Wp2,
    const float* __restrict__ bp2, float* __restrict__ out_fc)
{
    const int idx = blockIdx.x * 256 + threadIdx.x;   // 0..65535
    const int bi  = idx >> 11;
    const int c   = idx & 2047;
    float s = bp2[c];
    const float* hb = h2 + bi * HID_;
    for (int j = 0; j < HID_; ++j)
        s += hb[j] * Wp2[(size_t)j * RO_ + c];
    out_fc[idx] = s;
}

// ---------------------------------------------------------------------------
// Kernel D1: gpn_obj_ind int -> float copy (73728 elems)
// ---------------------------------------------------------------------------
__global__ __launch_bounds__(256) void gpn_copy_obj(
    const int* __restrict__ obj_ind, float* __restrict__ out)
{
    const int idx = blockIdx.x * 256 + threadIdx.x;
    out[idx] = (float)obj_ind[idx];
}

// ---------------------------------------------------------------------------
// Kernel D2: att_feats_out gather (32 x 36 x 1024)
// ---------------------------------------------------------------------------
__global__ __launch_bounds__(256) void gpn_gather_att(
    const int* __restrict__ obj_ind, const float* __restrict__ att,
    const int* __restrict__ ws_mi, float* __restrict__ out)
{
    const int idx = blockIdx.x * 256 + threadIdx.x;   // 0..1179647
    const int bi  = idx / (N_ * L_);
    const int rem = idx - bi * (N_ * L_);
    const int n   = rem / L_;
    const int l   = rem - n * L_;
    const int mi  = ws_mi[bi];
    const int o   = obj_ind[((size_t)bi * M_ + mi) * N_ + n];
    out[idx] = att[((size_t)bi * NMAX_ + o) * L_ + l];
}

extern "C" void kernel_launch(void* const* d_in, const int* in_sizes, int n_in,
                              void* d_out, int out_size, void* d_ws, size_t ws_size,
                              hipStream_t stream) {
    (void)in_sizes; (void)n_in; (void)out_size; (void)ws_size;
    // inputs (setup_inputs order): 0:b 1:N 2:K 3:L 4:gpn_obj_ind 5:gpn_pred_ind
    // 6:gpn_nrel_ind 7:gpn_pool_mtx 8:att_feats 9:x_pred 10:fc_feats
    // 11:att_masks 12:W1 13:b1 14:W2 15:b2 16:Wp1 17:bp1 18:Wp2 19:bp2
    const int*   obj_ind  = (const int*)d_in[4];
    const float* pool_mtx = (const float*)d_in[7];
    const float* att      = (const float*)d_in[8];
    const float* masks    = (const float*)d_in[11];
    const float* W1       = (const float*)d_in[12];
    const float* b1       = (const float*)d_in[13];
    const float* W2       = (const float*)d_in[14];
    const float* b2       = (const float*)d_in[15];
    const float* Wp1      = (const float*)d_in[16];
    const float* bp1      = (const float*)d_in[17];
    const float* Wp2      = (const float*)d_in[18];
    const float* bp2      = (const float*)d_in[19];

    // workspace layout
    char* ws = (char*)d_ws;
    float* ws_ro = (float*)ws;                                   // 2048*2048 f32
    float* ws_h  = (float*)(ws + (size_t)G_ * RO_ * 4);          // 2048*512 f32
    float* ws_h2 = (float*)(ws + (size_t)G_ * RO_ * 4 + (size_t)G_ * HID_ * 4); // 32*512
    int*   ws_mi = (int*)((char*)ws_h2 + (size_t)B_ * HID_ * 4); // 32 ints

    // output layout (floats, return-order concatenated)
    float* out        = (float*)d_out;
    float* out_submax = out;                       // 32
    float* out_obj    = out + 32;                  // 73728
    float* out_score  = out_obj + B_ * M_ * N_;    // 2048
    float* out_att    = out_score + G_;            // 1179648
    float* out_fc     = out_att + B_ * N_ * L_;    // 65536
    float* out_masks  = out_fc + B_ * RO_;         // 1152

    gpn_pool_gemm_reduce<<<G_, 1024, 0, stream>>>(obj_ind, pool_mtx, att, ws_ro);
    gpn_hidden_gemm<<<dim3(G_ / 32, HID_ / 128), 256, 0, stream>>>(ws_ro, W1, b1, ws_h);
    gpn_score<<<G_ / 8, 256, 0, stream>>>(ws_h, W2, b2, out_score);
    gpn_argmax<<<B_, 32, 0, stream>>>(out_score, masks, out_submax, out_masks, ws_mi);
    gpn_fc1<<<(B_ * HID_) / 256, 256, 0, stream>>>(ws_ro, ws_mi, Wp1, bp1, ws_h2);
    gpn_fc2<<<(B_ * RO_) / 256, 256, 0, stream>>>(ws_h2, Wp2, bp2, out_fc);
    gpn_copy_obj<<<(B_ * M_ * N_) / 256, 256, 0, stream>>>(obj_ind, out_obj);
    gpn_gather_att<<<(B_ * N_ * L_) / 256, 256, 0, stream>>>(obj_ind, att, ws_mi, out_att);
}